// TensorTrainLayer_57552561767068
// MI455X (gfx1250) — compile-verified
//
#include <hip/hip_runtime.h>
#include <hip/hip_bf16.h>

// CDNA5 (gfx1250) fp32 WMMA: D(16x16) = A(16x4) x B(4x16) + C, exact fp32.
typedef float v2f __attribute__((ext_vector_type(2)));
typedef float v8f __attribute__((ext_vector_type(8)));

__device__ __forceinline__ v8f wmma_f32(v2f a, v2f b, v8f c) {
    // 8 args: (neg_a, A, neg_b, B, c_mod, C, reuse_a, reuse_b)
    return __builtin_amdgcn_wmma_f32_16x16x4_f32(false, a, false, b, (short)0, c, false, false);
}

// LDS layout sizes (floats). Strides padded for bank-conflict-free B loads.
#define SK0_STRIDE 9     // step0 Kmat: 32 rows (o0*4+r1) x K=8 (m0)
#define SK_STRIDE 33     // steps1/2 Kmat: 32 rows x K=32 ; step3: 16 rows (8 zero)
#define INTER_M1_STRIDE 72   // (o0,r1,m1,x64): x-row stride 72 (pad 64->72)
#define INTER_R1_STRIDE 576  // 8*72
#define INTER_SLICE 2304     // per-o0 slice = 4*576
#define BUF_X_STRIDE 9       // step1 out (o1,r2,m2,x8): x-row stride 9 (pad 8->9)
#define BUF_R2_STRIDE 72     // 8*9
#define BUF_O1_STRIDE 288    // 4*72
#define B2_OA_STRIDE 36      // step2 out (oa'=64, r3, m3): oa stride 36 (pad 32->36)

__global__ __launch_bounds__(256, 1)
void tt_fwd_wmma_kernel(const float* __restrict__ x,
                        const float* __restrict__ c0,
                        const float* __restrict__ c1,
                        const float* __restrict__ c2,
                        const float* __restrict__ c3,
                        float* __restrict__ out,
                        int nrows)
{
    __shared__ float sK0[32 * SK0_STRIDE];      //  288
    __shared__ float sK1[32 * SK_STRIDE];       // 1056
    __shared__ float sK2[32 * SK_STRIDE];       // 1056
    __shared__ float sK3[16 * SK_STRIDE];       //  528 (rows 8..15 zero)
    __shared__ float sInter[32 * INTER_R1_STRIDE]; // 18432 (72KB)
    __shared__ float sBuf[8 * INTER_SLICE];        // 18432 (72KB), per-wave 2304

    const int tid  = threadIdx.x;
    const int lane = tid & 31;
    const int wave = tid >> 5;     // 8 waves; wave == o0 chunk id
    const int half = lane >> 4;    // lane group (K offset 0 / +2)
    const int lm   = lane & 15;    // M (A) / N (B,C,D) index within tile

    // ---- Build Kmats in WMMA-friendly row-major [M][K] layout ----
    // core shape (m, o, r, j); Kmat[M=o*rn+j][K=r*8+m]
    {
        int i = tid;                    // 256 elems, one per thread
        int M = i >> 3, K = i & 7;      // M=(o0*4+r1), K=m0 (r0==1)
        int o = M >> 2, j = M & 3;
        sK0[M * SK0_STRIDE + K] = c0[(K * 8 + o) * 4 + j];
    }
    for (int i = tid; i < 1024; i += 256) {
        int M = i >> 5, K = i & 31;
        int o = M >> 2, j = M & 3, r = K >> 3, m = K & 7;
        sK1[M * SK_STRIDE + K] = c1[((m * 8 + o) * 4 + r) * 4 + j];
        sK2[M * SK_STRIDE + K] = c2[((m * 8 + o) * 4 + r) * 4 + j];
    }
    for (int i = tid; i < 512; i += 256) {
        int M = i >> 5, K = i & 31;     // M=o3 (rnext==1)
        int r = K >> 3, m = K & 7;
        sK3[M * SK_STRIDE + K] = (M < 8) ? c3[(m * 8 + M) * 4 + r] : 0.0f;
    }
    __syncthreads();

    // ---- Hoist A fragments (cores) into registers; constant across rows ----
    // A element (Mrow, K): lane = Mrow%16 + 16*(K/2 within frag), vgpr = K&1
    v2f a0[2][2], a1[2][8], a2[2][8], a3[8];
    #pragma unroll
    for (int mt = 0; mt < 2; ++mt)
        #pragma unroll
        for (int kk = 0; kk < 2; ++kk) {
            const float* p = &sK0[(mt * 16 + lm) * SK0_STRIDE + kk * 4 + 2 * half];
            v2f a; a.x = p[0]; a.y = p[1]; a0[mt][kk] = a;
        }
    #pragma unroll
    for (int mt = 0; mt < 2; ++mt)
        #pragma unroll
        for (int kk = 0; kk < 8; ++kk) {
            const float* p1 = &sK1[(mt * 16 + lm) * SK_STRIDE + kk * 4 + 2 * half];
            v2f a; a.x = p1[0]; a.y = p1[1]; a1[mt][kk] = a;
            const float* p2 = &sK2[(mt * 16 + lm) * SK_STRIDE + kk * 4 + 2 * half];
            v2f b; b.x = p2[0]; b.y = p2[1]; a2[mt][kk] = b;
        }
    #pragma unroll
    for (int kk = 0; kk < 8; ++kk) {
        const float* p = &sK3[lm * SK_STRIDE + kk * 4 + 2 * half];
        v2f a; a.x = p[0]; a.y = p[1]; a3[kk] = a;
    }

    float* myBuf = &sBuf[wave * INTER_SLICE];     // step1 output
    float* myB2  = &sInter[wave * INTER_SLICE];   // step2 output reuses own inter slice

    for (int row = blockIdx.x; row < nrows; row += gridDim.x) {
        const float* xr = x + (size_t)row * 4096;

        // ================= Step 0: M=32 (o0*4+r1), K=8 (m0), N=512 (x) ======
        // This wave owns x in [wave*64, wave*64+64). B read from global:
        // B[(k=m0)][x] = xr[k*512 + x].
        #pragma unroll
        for (int nt = 0; nt < 4; ++nt) {
            int xg = wave * 64 + nt * 16 + lm;
            v2f b[2];
            #pragma unroll
            for (int kk = 0; kk < 2; ++kk) {
                int K = kk * 4 + 2 * half;
                v2f t;
                t.x = xr[(size_t)K * 512 + xg];
                t.y = xr[(size_t)(K + 1) * 512 + xg];
                b[kk] = t;
            }
            v8f acc0 = {}; v8f acc1 = {};
            #pragma unroll
            for (int kk = 0; kk < 2; ++kk) {
                acc0 = wmma_f32(a0[0][kk], b[kk], acc0);
                acc1 = wmma_f32(a0[1][kk], b[kk], acc1);
            }
            // Store (o0,r1,m1,x64) padded: idx = Mg*576 + m1*72 + xr64
            int m1 = xg >> 6, x64 = xg & 63;
            #pragma unroll
            for (int v = 0; v < 8; ++v) {
                int Mg0 = 8 * half + v;
                sInter[Mg0 * INTER_R1_STRIDE + m1 * INTER_M1_STRIDE + x64] = acc0[v];
                int Mg1 = 16 + 8 * half + v;
                sInter[Mg1 * INTER_R1_STRIDE + m1 * INTER_M1_STRIDE + x64] = acc1[v];
            }
        }
        __syncthreads();   // all waves wrote every o0 slice

        // ===== Steps 1-3 are wave-private on the o0==wave slice (no barriers) =====
        const float* src1 = &sInter[wave * INTER_SLICE];  // B addr = k*72 + x

        // Step 1: M=32 (o1*4+r2), K=32 (r1*8+m1), N=64 (x)
        #pragma unroll
        for (int nt = 0; nt < 4; ++nt) {
            int xg = nt * 16 + lm;
            v8f acc0 = {}, acc1 = {};
            #pragma unroll
            for (int kk = 0; kk < 8; ++kk) {
                int K = kk * 4 + 2 * half;
                v2f b;
                b.x = src1[K * INTER_M1_STRIDE + xg];
                b.y = src1[(K + 1) * INTER_M1_STRIDE + xg];
                acc0 = wmma_f32(a1[0][kk], b, acc0);
                acc1 = wmma_f32(a1[1][kk], b, acc1);
            }
            // Store (o1,r2,m2=x>>3,x8) padded
            int xo = (xg >> 3) * BUF_X_STRIDE + (xg & 7);
            #pragma unroll
            for (int v = 0; v < 8; ++v) {
                int Mg0 = 8 * half + v;
                myBuf[(Mg0 >> 2) * BUF_O1_STRIDE + (Mg0 & 3) * BUF_R2_STRIDE + xo] = acc0[v];
                int Mg1 = 16 + 8 * half + v;
                myBuf[(Mg1 >> 2) * BUF_O1_STRIDE + (Mg1 & 3) * BUF_R2_STRIDE + xo] = acc1[v];
            }
        }

        // Step 2: M=32 (o2*4+r3), K=32 (r2*8+m2), free=(o1=8, x=8) -> N packs 2 o1 per tile
        #pragma unroll
        for (int nt = 0; nt < 4; ++nt) {
            int oa = nt * 2 + (lm >> 3);   // o1
            int x8 = lm & 7;
            v8f acc0 = {}, acc1 = {};
            const float* p = &myBuf[oa * BUF_O1_STRIDE + x8];  // + k*9
            #pragma unroll
            for (int kk = 0; kk < 8; ++kk) {
                int K = kk * 4 + 2 * half;
                v2f b;
                b.x = p[K * BUF_X_STRIDE];
                b.y = p[(K + 1) * BUF_X_STRIDE];
                acc0 = wmma_f32(a2[0][kk], b, acc0);
                acc1 = wmma_f32(a2[1][kk], b, acc1);
            }
            // Store (oa'=o1*8+o2)*36 + r3*8 + x8  (reuses own inter slice)
            #pragma unroll
            for (int v = 0; v < 8; ++v) {
                int Mg0 = 8 * half + v;
                myB2[(oa * 8 + (Mg0 >> 2)) * B2_OA_STRIDE + (Mg0 & 3) * 8 + x8] = acc0[v];
                int Mg1 = 16 + 8 * half + v;
                myB2[(oa * 8 + (Mg1 >> 2)) * B2_OA_STRIDE + (Mg1 & 3) * 8 + x8] = acc1[v];
            }
        }

        // Step 3: M=8 (o3, rows 8..15 zero), K=32 (r3*8+m3), N=64 (oa'=o1*8+o2)
        float* outw = out + (size_t)row * 4096 + wave * 512;
        #pragma unroll
        for (int nt = 0; nt < 4; ++nt) {
            int n = nt * 16 + lm;          // oa'
            v8f acc = {};
            const float* p = &myB2[n * B2_OA_STRIDE];
            #pragma unroll
            for (int kk = 0; kk < 8; ++kk) {
                int K = kk * 4 + 2 * half;
                v2f b;
                b.x = p[K];
                b.y = p[K + 1];
                acc = wmma_f32(a3[kk], b, acc);
            }
            // D rows 0..7 (lanes 0-15) valid; out idx = o1*64 + o2*8 + o3 = n*8 + v
            if (half == 0) {
                float4 lo = make_float4(acc[0], acc[1], acc[2], acc[3]);
                float4 hi = make_float4(acc[4], acc[5], acc[6], acc[7]);
                float* q = outw + n * 8;
                *(float4*)(q)     = lo;
                *(float4*)(q + 4) = hi;
            }
        }
        __syncthreads();   // before next row's step0 overwrites sInter
    }
}

extern "C" void kernel_launch(void* const* d_in, const int* in_sizes, int n_in,
                              void* d_out, int out_size, void* d_ws, size_t ws_size,
                              hipStream_t stream) {
    (void)in_sizes; (void)n_in; (void)out_size; (void)d_ws; (void)ws_size;
    const float* x  = (const float*)d_in[0];
    const float* c0 = (const float*)d_in[1];
    const float* c1 = (const float*)d_in[2];
    const float* c2 = (const float*)d_in[3];
    const float* c3 = (const float*)d_in[4];
    float* out = (float*)d_out;
    // 4096 rows; 1024 persistent-ish blocks, 4 rows each (grid-stride).
    hipLaunchKernelGGL(tt_fwd_wmma_kernel, dim3(1024), dim3(256), 0, stream,
                       x, c0, c1, c2, c3, out, 4096);
}